// SSARLSTM_21921513078898
// MI455X (gfx1250) — compile-verified
//
#include <hip/hip_runtime.h>
#include <hip/hip_bf16.h>
#include <math.h>

// ---------------------------------------------------------------------------
// 4-layer LSTM (B=64, T=512, H=I=256) + FC(256->60), fp32 via WMMA f32 16x16x4
// Workspace layout (floats):
//   ybuf : B*T*H            = 8,388,608
//   wpk  : 4 * 512*1024     = 2,097,152   (packed concat [W_ih;W_hh]^T per layer)
//   fcpk : 256*64           = 16,384      (packed fc_w^T, N padded 60->64)
// total ~40 MB.
// ---------------------------------------------------------------------------

typedef __attribute__((ext_vector_type(2))) float v2f;
typedef __attribute__((ext_vector_type(8))) float v8f;

#define HDIM 256
#define TLEN 512
#define BATCH 64
#define GDIM 1024   // 4*H
#define KDIM 512    // I + H
#define CDIM 60
#define AST  520    // LDS A-buffer row stride (512 + 8 pad, bank-conflict relief)

__device__ __forceinline__ float sigmoidf_(float x) {
    return 1.0f / (1.0f + __expf(-x));
}
__device__ __forceinline__ float tanhf_(float x) {
    // robust: saturates to +-1 for large |x| (no inf/inf)
    return 1.0f - 2.0f / (__expf(2.0f * x) + 1.0f);
}

#define WMMA_F32X4(A, B, C) \
    __builtin_amdgcn_wmma_f32_16x16x4_f32(false, (A), false, (B), (short)0, (C), false, false)

// -------- one-time weight packing --------------------------------------------
// element (k, n) of concat W^T (k in [0,512), n in [0,1024)) stored at
//   (k>>1)*2048 + n*2 + (k&1)
// so a lane's two consecutive-k B values for column n are one aligned b64 load.
__global__ __launch_bounds__(256) void pack_lstm_weights(
    const float* __restrict__ Wih0, const float* __restrict__ Whh0,
    const float* __restrict__ Wih1, const float* __restrict__ Whh1,
    const float* __restrict__ Wih2, const float* __restrict__ Whh2,
    const float* __restrict__ Wih3, const float* __restrict__ Whh3,
    float* __restrict__ wpk) {
    int gid = blockIdx.x * blockDim.x + threadIdx.x;
    if (gid >= 4 * KDIM * GDIM) return;
    int l   = gid >> 19;          // / (512*1024)
    int rem = gid & (KDIM * GDIM - 1);
    int k   = rem >> 10;
    int n   = rem & 1023;
    const float* Wih = (l == 0) ? Wih0 : (l == 1) ? Wih1 : (l == 2) ? Wih2 : Wih3;
    const float* Whh = (l == 0) ? Whh0 : (l == 1) ? Whh1 : (l == 2) ? Whh2 : Whh3;
    float v = (k < HDIM) ? Wih[n * HDIM + k] : Whh[n * HDIM + (k - HDIM)];
    wpk[(size_t)l * (KDIM * GDIM) + (size_t)(k >> 1) * (2 * GDIM) + n * 2 + (k & 1)] = v;
}

__global__ __launch_bounds__(256) void pack_fc(const float* __restrict__ fcw,
                                               float* __restrict__ fcpk) {
    int gid = blockIdx.x * blockDim.x + threadIdx.x;   // 256*64
    if (gid >= HDIM * 64) return;
    int k = gid >> 6, n = gid & 63;
    float v = (n < CDIM) ? fcw[n * HDIM + k] : 0.0f;
    fcpk[(k >> 1) * 128 + n * 2 + (k & 1)] = v;
}

// -------- persistent per-layer LSTM scan -------------------------------------
// grid = 4 workgroups (one per 16-batch tile), block = 512 threads (16 waves).
// Wave w owns hidden units j = w*16 + (lane&15) and computes the four matching
// gate tiles (N = gate*256 + w*16); cell state c stays in registers.
__global__ __launch_bounds__(512) void lstm_scan(
    const float* xin, float* yout,              // may alias (in-place per layer)
    const float* __restrict__ wpk,
    const float* __restrict__ b_ih, const float* __restrict__ b_hh,
    float* __restrict__ hT, float* __restrict__ cT) {
    __shared__ float abuf[16 * AST];            // rows=batch; cols 0..255=x_t, 256..511=h

    const int tid   = threadIdx.x;
    const int wave  = tid >> 5;
    const int lane  = tid & 31;
    const int b0    = blockIdx.x * 16;
    const int arow  = lane & 15;                // A-matrix M (batch row in tile)
    const int khalf = lane >> 4;                // 0: K={k,k+1}, 1: K={k+2,k+3}
    const int jcol  = (wave << 4) + (lane & 15);// owned hidden unit / B-matrix N

    for (int i = tid; i < 16 * AST; i += 512) abuf[i] = 0.0f;   // zero h (and x)

    const float bi = b_ih[jcol]       + b_hh[jcol];
    const float bf = b_ih[256 + jcol] + b_hh[256 + jcol];
    const float bg = b_ih[512 + jcol] + b_hh[512 + jcol];
    const float bo = b_ih[768 + jcol] + b_hh[768 + jcol];

    float creg[8], hreg[8];
#pragma unroll
    for (int r = 0; r < 8; ++r) { creg[r] = 0.0f; hreg[r] = 0.0f; }

    // x_t staging: wave w loads batch row w, lane loads 8 consecutive floats
    const float* xrow_ptr  = xin + (size_t)(b0 + wave) * (TLEN * HDIM);
    float*       yrow_base = yout + (size_t)b0 * (TLEN * HDIM);

    for (int t = 0; t < TLEN; ++t) {
        {   // stage x_t into abuf cols [0,256)
            const float4* src = (const float4*)(xrow_ptr + t * HDIM + (lane << 3));
            float4 v0 = src[0], v1 = src[1];
            float4* dst = (float4*)&abuf[wave * AST + (lane << 3)];
            dst[0] = v0; dst[1] = v1;
        }
        __syncthreads();    // x_t + previous-step h visible to everyone

        v8f acc0 = {0, 0, 0, 0, 0, 0, 0, 0};
        v8f acc1 = {0, 0, 0, 0, 0, 0, 0, 0};
        v8f acc2 = {0, 0, 0, 0, 0, 0, 0, 0};
        v8f acc3 = {0, 0, 0, 0, 0, 0, 0, 0};

        const float* arowp = &abuf[arow * AST + 2 * khalf];
        for (int k0 = 0; k0 < KDIM; k0 += 4) {
            v2f a = *(const v2f*)(arowp + k0);                       // ds_load_b64
            const float* wrow =
                wpk + ((size_t)((k0 >> 1) + khalf)) * (2 * GDIM) + 2 * jcol;
            v2f bI = *(const v2f*)(wrow);          // gate i : N = jcol
            v2f bF = *(const v2f*)(wrow + 512);    // gate f : N = 256 + jcol
            v2f bG = *(const v2f*)(wrow + 1024);   // gate g : N = 512 + jcol
            v2f bO = *(const v2f*)(wrow + 1536);   // gate o : N = 768 + jcol
            acc0 = WMMA_F32X4(a, bI, acc0);
            acc1 = WMMA_F32X4(a, bF, acc1);
            acc2 = WMMA_F32X4(a, bG, acc2);
            acc3 = WMMA_F32X4(a, bO, acc3);
        }
        __syncthreads();    // all GEMM reads of abuf done before h overwrite

        // gate nonlinearity + cell update: pure register math, identical (b,j)
        // mapping across the four accumulators (C-layout: M = r + 8*khalf).
#pragma unroll
        for (int r = 0; r < 8; ++r) {
            float iv = sigmoidf_(acc0[r] + bi);
            float fv = sigmoidf_(acc1[r] + bf);
            float gv = tanhf_(acc2[r] + bg);
            float ov = sigmoidf_(acc3[r] + bo);
            float cn = fv * creg[r] + iv * gv;
            float hn = ov * tanhf_(cn);
            creg[r] = cn;
            hreg[r] = hn;
            int brow = r + (khalf << 3);
            abuf[brow * AST + 256 + jcol] = hn;                       // h for t+1
            yrow_base[(size_t)brow * (TLEN * HDIM) + t * HDIM + jcol] = hn;
        }
        // no barrier: next x-stage writes cols [0,256) — disjoint from h writes;
        // next GEMM is gated by the top-of-loop __syncthreads().
    }

    // final h_T / c_T straight from registers
#pragma unroll
    for (int r = 0; r < 8; ++r) {
        int brow = r + (khalf << 3);
        hT[(size_t)(b0 + brow) * HDIM + jcol] = hreg[r];
        cT[(size_t)(b0 + brow) * HDIM + jcol] = creg[r];
    }
}

// -------- final FC + permute: label[b][c][t] = y[b][t][:] . fc_w[c][:] + fc_b -
// one wave per (batch, 16-row t-tile); 4 N-tiles cover padded C=64.
__global__ __launch_bounds__(256) void fc_kernel(
    const float* __restrict__ y, const float* __restrict__ fcpk,
    const float* __restrict__ fcb, float* __restrict__ label) {
    const int tid   = threadIdx.x;
    const int lane  = tid & 31;
    const int gw    = blockIdx.x * 8 + (tid >> 5);   // 0..2047
    const int b     = gw >> 5;
    const int t0    = (gw & 31) << 4;
    const int khalf = lane >> 4;
    const int trow  = t0 + (lane & 15);

    const float* arowp = y + (size_t)b * (TLEN * HDIM) + (size_t)trow * HDIM + 2 * khalf;

    v8f acc[4];
#pragma unroll
    for (int j = 0; j < 4; ++j) acc[j] = (v8f){0, 0, 0, 0, 0, 0, 0, 0};

    for (int k0 = 0; k0 < HDIM; k0 += 4) {
        v2f a = *(const v2f*)(arowp + k0);
        const float* wrow = fcpk + ((k0 >> 1) + khalf) * 128 + 2 * (lane & 15);
#pragma unroll
        for (int j = 0; j < 4; ++j) {
            v2f bv = *(const v2f*)(wrow + 32 * j);   // N-tile base = j*16
            acc[j] = WMMA_F32X4(a, bv, acc[j]);
        }
    }

#pragma unroll
    for (int j = 0; j < 4; ++j) {
        int n = (j << 4) + (lane & 15);
        if (n < CDIM) {
            float bias = fcb[n];
            float* outp =
                label + (size_t)b * (CDIM * TLEN) + (size_t)n * TLEN + t0 + (khalf << 3);
#pragma unroll
            for (int r = 0; r < 8; ++r) outp[r] = acc[j][r] + bias;
        }
    }
}

// ---------------------------------------------------------------------------
extern "C" void kernel_launch(void* const* d_in, const int* in_sizes, int n_in,
                              void* d_out, int out_size, void* d_ws, size_t ws_size,
                              hipStream_t stream) {
    const float* x = (const float*)d_in[0];
    const float *Wih[4], *Whh[4], *bih[4], *bhh[4];
    for (int l = 0; l < 4; ++l) {
        Wih[l] = (const float*)d_in[1 + 4 * l];
        Whh[l] = (const float*)d_in[2 + 4 * l];
        bih[l] = (const float*)d_in[3 + 4 * l];
        bhh[l] = (const float*)d_in[4 + 4 * l];
    }
    const float* fcw = (const float*)d_in[17];
    const float* fcb = (const float*)d_in[18];

    float* out   = (float*)d_out;
    float* label = out;                                 // [64,60,512]
    float* hs    = out + (size_t)BATCH * CDIM * TLEN;   // [4,64,256]
    float* cs    = hs + (size_t)4 * BATCH * HDIM;       // [4,64,256]

    float* ws   = (float*)d_ws;                         // needs ~40.1 MB
    float* ybuf = ws;                                   // 8,388,608 floats
    float* wpk  = ybuf + (size_t)BATCH * TLEN * HDIM;   // 2,097,152 floats
    float* fcpk = wpk + (size_t)4 * KDIM * GDIM;        // 16,384 floats

    pack_lstm_weights<<<(4 * KDIM * GDIM) / 256, 256, 0, stream>>>(
        Wih[0], Whh[0], Wih[1], Whh[1], Wih[2], Whh[2], Wih[3], Whh[3], wpk);
    pack_fc<<<(HDIM * 64) / 256, 256, 0, stream>>>(fcw, fcpk);

    for (int l = 0; l < 4; ++l) {
        lstm_scan<<<4, 512, 0, stream>>>(
            (l == 0) ? x : ybuf, ybuf, wpk + (size_t)l * KDIM * GDIM,
            bih[l], bhh[l],
            hs + (size_t)l * BATCH * HDIM, cs + (size_t)l * BATCH * HDIM);
    }

    fc_kernel<<<256, 256, 0, stream>>>(ybuf, fcpk, fcb, label);
}